// MolConv_51049981280383
// MI455X (gfx1250) — compile-verified
//
#include <hip/hip_runtime.h>

// MolConv: deg/norm -> edge scatter-sum of concat(efeat, feat[src]) -> fused
// (sum_f*norm) @ W[48,64] * norm + bias, relu.
// Phase 2 (scatter) is the HBM-bound phase (~0.65 GB traffic ~ 28us @ 23.3TB/s).
// Phase 3 GEMM (0.6 GFLOP) runs on V_WMMA_F32_16X16X4_F32 in full fp32.

#define N_NODES 100000
#define N_EDGES 1600000
#define NODE_F  32
#define EDGE_F  16
#define IN_F    48     // EDGE_F + NODE_F
#define OUT_F   64

typedef __attribute__((ext_vector_type(2))) float v2f;
typedef __attribute__((ext_vector_type(8))) float v8f;

// ---------------------------------------------------------------- K0: zero ws
__global__ void molconv_zero_kernel(float* __restrict__ ws, int n) {
  int i = blockIdx.x * blockDim.x + threadIdx.x;
  if (i < n) ws[i] = 0.0f;
}

// ------------------------------------------------- K1: edge scatter (atomics)
// One thread per (edge, feature-slot). Adjacent threads hit adjacent feature
// slots of the same destination row -> atomic traffic coalesces per cacheline.
__global__ void molconv_scatter_kernel(const float* __restrict__ feat,
                                       const float* __restrict__ efeat,
                                       const int*   __restrict__ src,
                                       const int*   __restrict__ dst,
                                       float* __restrict__ sum_f,
                                       float* __restrict__ deg) {
  long long idx = (long long)blockIdx.x * blockDim.x + threadIdx.x;
  const long long total = (long long)N_EDGES * IN_F;
  if (idx >= total) return;
  int e = (int)(idx / IN_F);
  int j = (int)(idx - (long long)e * IN_F);
  int d = dst[e];
  float val;
  if (j < EDGE_F) {
    val = efeat[(long long)e * EDGE_F + j];
  } else {
    val = feat[(long long)src[e] * NODE_F + (j - EDGE_F)];
  }
  atomicAdd(&sum_f[(long long)d * IN_F + j], val);
  if (j == 0) atomicAdd(&deg[d], 1.0f);
}

// --------------------------------------------- K2: WMMA GEMM + norm/bias/relu
// 8 wave32 per block; each wave computes a 16x64 output tile:
//   D[16,64] = (sum_f[16 rows] * norm) @ W[48,64]
// via 12 K-steps x 4 N-tiles of v_wmma_f32_16x16x4_f32.
__global__ __launch_bounds__(256)
void molconv_gemm_kernel(const float* __restrict__ sum_f,
                         const float* __restrict__ deg,
                         const float* __restrict__ weight,  // [IN_F][OUT_F]
                         const float* __restrict__ bias,    // [OUT_F]
                         float* __restrict__ out) {         // [N][OUT_F]
  __shared__ float wsh[IN_F * OUT_F];  // 12 KB, shared by all 8 waves
  const int tid = threadIdx.x;
  for (int i = tid; i < IN_F * OUT_F; i += 256) wsh[i] = weight[i];
  __syncthreads();

  const int wave  = tid >> 5;
  const int lane  = tid & 31;
  const int base  = (blockIdx.x * 8 + wave) * 16;   // first node row of tile
  if (base >= N_NODES) return;                      // N_NODES % 16 == 0

  const int l15   = lane & 15;
  const int khalf = lane >> 4;                      // 0: K+0/1, 1: K+2/3

  // A-fragment row for this lane (16x4 f32 A layout: lanes 0-15 & 16-31 both
  // cover M=0..15; VGPR0/1 hold K pairs split by lane half).
  const int arow = base + l15;
  const float nrmA = rsqrtf(fmaxf(deg[arow], 1.0f));
  const float* aptr = sum_f + (long long)arow * IN_F + 2 * khalf;

  v8f acc[4];
  acc[0] = v8f{0.f,0.f,0.f,0.f,0.f,0.f,0.f,0.f};
  acc[1] = acc[0]; acc[2] = acc[0]; acc[3] = acc[0];

#pragma unroll
  for (int k0 = 0; k0 < IN_F; k0 += 4) {
    v2f a;
    a.x = aptr[k0]     * nrmA;   // K = k0 + 2*khalf
    a.y = aptr[k0 + 1] * nrmA;   // K = k0 + 2*khalf + 1
    const int kb = k0 + 2 * khalf;
#pragma unroll
    for (int n = 0; n < 4; ++n) {
      v2f b;                     // B 4x16: same K split by lane half
      b.x = wsh[kb       * OUT_F + n * 16 + l15];
      b.y = wsh[(kb + 1) * OUT_F + n * 16 + l15];
      acc[n] = __builtin_amdgcn_wmma_f32_16x16x4_f32(
          /*neg_a=*/false, a, /*neg_b=*/false, b,
          /*c_mod=*/(short)0, acc[n], /*reuse_a=*/false, /*reuse_b=*/false);
    }
  }

  // Epilogue: C/D layout -> VGPR r holds M=r (lanes 0-15) / M=r+8 (lanes 16-31)
#pragma unroll
  for (int r = 0; r < 8; ++r) {
    const int row = base + r + 8 * khalf;
    const float nr = rsqrtf(fmaxf(deg[row], 1.0f));
#pragma unroll
    for (int n = 0; n < 4; ++n) {
      const int col = n * 16 + l15;
      const float v = acc[n][r] * nr + bias[col];
      out[(long long)row * OUT_F + col] = fmaxf(v, 0.0f);
    }
  }
}

// ---------------------------------------------------------------------- launch
extern "C" void kernel_launch(void* const* d_in, const int* in_sizes, int n_in,
                              void* d_out, int out_size, void* d_ws, size_t ws_size,
                              hipStream_t stream) {
  const float* feat   = (const float*)d_in[0];
  const float* efeat  = (const float*)d_in[1];
  const int*   src    = (const int*)  d_in[2];
  const int*   dst    = (const int*)  d_in[3];
  const float* weight = (const float*)d_in[4];
  const float* bias   = (const float*)d_in[5];
  float*       out    = (float*)d_out;

  float* sum_f = (float*)d_ws;                        // [N_NODES][IN_F]
  float* deg   = sum_f + (size_t)N_NODES * IN_F;      // [N_NODES]

  // K0: zero sum_f + deg
  const int nzero = N_NODES * (IN_F + 1);
  molconv_zero_kernel<<<(nzero + 255) / 256, 256, 0, stream>>>((float*)d_ws, nzero);

  // K1: scatter-sum messages + degree
  const long long nscatter = (long long)N_EDGES * IN_F;
  const int sblocks = (int)((nscatter + 255) / 256);
  molconv_scatter_kernel<<<sblocks, 256, 0, stream>>>(feat, efeat, src, dst,
                                                      sum_f, deg);

  // K2: WMMA GEMM + epilogue. 6250 row-tiles, 8 per block.
  const int ntiles  = N_NODES / 16;            // 6250
  const int gblocks = (ntiles + 7) / 8;        // 782
  molconv_gemm_kernel<<<gblocks, 256, 0, stream>>>(sum_f, deg, weight, bias, out);
}